// AutoregressiveModel_86861418594880
// MI455X (gfx1250) — compile-verified
//
#include <hip/hip_runtime.h>
#include <hip/hip_bf16.h>

// ---------------------------------------------------------------------------
// Lattice causal-graph GNN for MI455X (gfx1250, wave32).
//   layer1: (2->128) per-type project + scatter-add (no self)
//   LN+tanh
//   layer2: (128->128) per-type WMMA f32 16x16x4 GEMM (32x32 tile/wave) + scatter
//   LN+tanh
//   layer3: (128->2) per-type wave-dot project + scatter-add -> d_out
// ---------------------------------------------------------------------------

#define BATCH   32
#define NODES   1024
#define ROWS    (BATCH * NODES)   // 32768
#define FDIM    128
#define NTYPES  6

typedef __attribute__((ext_vector_type(2))) float v2f;
typedef __attribute__((ext_vector_type(8))) float v8f;

// ---------------------------------------------------------------- zero fill
__global__ void zero_f32(float* __restrict__ p, long long n) {
  long long i = (long long)blockIdx.x * blockDim.x + threadIdx.x;
  long long stride = (long long)gridDim.x * blockDim.x;
  for (; i < n; i += stride) p[i] = 0.0f;
}

// ------------------------------------------------- layer1 projection (2->128)
// P[r, f] = X[r,0]*W[0,f] + X[r,1]*W[1,f] + b[f]
__global__ void proj_fin2(const float* __restrict__ X, const float* __restrict__ W,
                          const float* __restrict__ bias, float* __restrict__ P) {
  long long idx = (long long)blockIdx.x * blockDim.x + threadIdx.x;
  if (idx >= (long long)ROWS * FDIM) return;
  int f = (int)(idx & (FDIM - 1));
  long long r = idx >> 7;
  float x0 = X[r * 2 + 0];
  float x1 = X[r * 2 + 1];
  P[idx] = fmaf(x0, W[f], fmaf(x1, W[FDIM + f], bias[f]));
}

// --------------------------------------- layer2 WMMA GEMM: P = H*W + b (f32)
// H: ROWS x 128 row-major, W: 128 x 128 row-major, P: ROWS x 128.
// One wave computes a 32x32 tile (2x2 grid of 16x16 WMMA tiles) so each
// A/B fragment feeds two V_WMMA_F32_16X16X4_F32 issues: 8 B/lane per WMMA
// instead of 16, doubling arithmetic intensity against L2.
// Fragment layouts per CDNA5 ISA 7.12.2:
//   A (16x4): lanes 0-15 hold {A[m][k],A[m][k+1]}, lanes 16-31 {k+2,k+3}
//   B (4x16): VGPR0 = row k (+2 for hi lanes), VGPR1 = row k+1 (+2 for hi)
//   C (16x16): VGPR j -> row j (lanes 0-15) / j+8 (lanes 16-31), col = lane&15
__global__ void gemm128_wmma_f32(const float* __restrict__ H,
                                 const float* __restrict__ W,
                                 const float* __restrict__ bias,
                                 float* __restrict__ P) {
  const int lane = threadIdx.x;                      // 0..31
  const int tile = blockIdx.x * blockDim.y + threadIdx.y;
  const int tm = (tile >> 2) * 32;                   // 128/32 = 4 N-tiles
  const int tn = (tile & 3) * 32;
  const int l15 = lane & 15;
  const int hi  = lane >> 4;                         // 0 | 1

  const float* arow0 = H + (long long)(tm + l15) * FDIM;       // rows tm..tm+15
  const float* arow1 = arow0 + 16LL * FDIM;                    // rows tm+16..+31
  const float* wc    = W + tn + l15;                           // col base
  __builtin_prefetch(arow0, 0, 0);                   // global_prefetch_b8
  __builtin_prefetch(arow1, 0, 0);

  v8f c00 = {}, c01 = {}, c10 = {}, c11 = {};
  #pragma unroll 4
  for (int k = 0; k < FDIM; k += 4) {
    v2f a0 = *(const v2f*)(arow0 + k + 2 * hi);      // 8B-aligned b64 loads
    v2f a1 = *(const v2f*)(arow1 + k + 2 * hi);
    v2f b0, b1;
    const float* wr = wc + (long long)(k + 2 * hi) * FDIM;
    b0.x = wr[0];         b0.y = wr[FDIM];           // cols tn..tn+15
    b1.x = wr[16];        b1.y = wr[FDIM + 16];      // cols tn+16..tn+31
    c00 = __builtin_amdgcn_wmma_f32_16x16x4_f32(false, a0, false, b0, (short)0, c00, false, false);
    c01 = __builtin_amdgcn_wmma_f32_16x16x4_f32(false, a0, false, b1, (short)0, c01, false, false);
    c10 = __builtin_amdgcn_wmma_f32_16x16x4_f32(false, a1, false, b0, (short)0, c10, false, false);
    c11 = __builtin_amdgcn_wmma_f32_16x16x4_f32(false, a1, false, b1, (short)0, c11, false, false);
  }

  float bv0 = bias[tn + l15];
  float bv1 = bias[tn + 16 + l15];
  // C layout: VGPR j -> row (tm + 8*hi + j), col tn + l15 (+16 for right block)
  float* o0 = P + (long long)(tm + 8 * hi) * FDIM + tn + l15;  // upper 16 rows
  float* o1 = o0 + 16LL * FDIM;                                // lower 16 rows
  #pragma unroll
  for (int j = 0; j < 8; ++j) {
    o0[(long long)j * FDIM +  0] = c00[j] + bv0;
    o0[(long long)j * FDIM + 16] = c01[j] + bv1;
    o1[(long long)j * FDIM +  0] = c10[j] + bv0;
    o1[(long long)j * FDIM + 16] = c11[j] + bv1;
  }
}

// ------------------------------------------------ layer3 projection (128->2)
// one wave per row: P[r,0..1] = dot(H[r,:], W[:,j]) + b[j]
__global__ void proj_fout2(const float* __restrict__ H, const float* __restrict__ W,
                           const float* __restrict__ bias, float* __restrict__ P) {
  int row = blockIdx.x * blockDim.y + threadIdx.y;
  if (row >= ROWS) return;
  int lane = threadIdx.x;
  const float* hp = H + (long long)row * FDIM;
  float a0 = 0.0f, a1 = 0.0f;
  #pragma unroll
  for (int i = 0; i < 4; ++i) {
    int f = lane + 32 * i;
    float hv = hp[f];
    a0 = fmaf(hv, W[f * 2 + 0], a0);
    a1 = fmaf(hv, W[f * 2 + 1], a1);
  }
  #pragma unroll
  for (int off = 16; off; off >>= 1) {
    a0 += __shfl_xor(a0, off, 32);
    a1 += __shfl_xor(a1, off, 32);
  }
  if (lane == 0) {
    P[(long long)row * 2 + 0] = a0 + bias[0];
    P[(long long)row * 2 + 1] = a1 + bias[1];
  }
}

// -------------------------------------------------------------- scatter add
// out[b, t[e], f] += P[b, s[e], f]   (global_atomic_add_f32)
__global__ void scatter_add(const float* __restrict__ P, const int* __restrict__ s,
                            const int* __restrict__ t, float* __restrict__ out,
                            long long n_edges, int F) {
  long long total = n_edges * BATCH * F;
  long long stride = (long long)gridDim.x * blockDim.x;
  for (long long idx = (long long)blockIdx.x * blockDim.x + threadIdx.x;
       idx < total; idx += stride) {
    int f = (int)(idx % F);
    long long r = idx / F;
    int b = (int)(r % BATCH);
    long long e = r / BATCH;
    int ss = s[e], tt = t[e];
    atomicAdd(&out[((long long)b * NODES + tt) * F + f],
              P[((long long)b * NODES + ss) * F + f]);
  }
}

// ----------------------------------------------- fused LayerNorm + tanh (128)
__global__ void ln_tanh(float* __restrict__ h, const float* __restrict__ g,
                        const float* __restrict__ bta) {
  int row = blockIdx.x * blockDim.y + threadIdx.y;
  if (row >= ROWS) return;
  int lane = threadIdx.x;
  float* p = h + (long long)row * FDIM;
  float v[4];
  float sum = 0.0f;
  #pragma unroll
  for (int i = 0; i < 4; ++i) { v[i] = p[lane + 32 * i]; sum += v[i]; }
  #pragma unroll
  for (int off = 16; off; off >>= 1) sum += __shfl_xor(sum, off, 32);
  float mu = sum * (1.0f / FDIM);
  float vs = 0.0f;
  #pragma unroll
  for (int i = 0; i < 4; ++i) { float d = v[i] - mu; vs = fmaf(d, d, vs); }
  #pragma unroll
  for (int off = 16; off; off >>= 1) vs += __shfl_xor(vs, off, 32);
  float rs = rsqrtf(vs * (1.0f / FDIM) + 1e-5f);
  #pragma unroll
  for (int i = 0; i < 4; ++i) {
    int f = lane + 32 * i;
    p[f] = tanhf(fmaf((v[i] - mu) * rs, g[f], bta[f]));
  }
}

// ---------------------------------------------------------------------------
extern "C" void kernel_launch(void* const* d_in, const int* in_sizes, int n_in,
                              void* d_out, int out_size, void* d_ws, size_t ws_size,
                              hipStream_t stream) {
  // Input layout (setup_inputs() insertion order):
  //  [0]                x (32,1024,2)
  //  [1 + 12*l + 2*i]   gc[l][type_i].W      (l=0..2, i=0..5 in TYPES order)
  //  [1 + 12*l + 2*i+1] gc[l][type_i].b
  //  [37],[38]          ln0.g, ln0.b     [39],[40] ln1.g, ln1.b
  //  [41 + 2*i]         edges[type_i].s  [42 + 2*i] edges[type_i].t
  const float* x = (const float*)d_in[0];
  auto Wp = [&](int l, int i) { return (const float*)d_in[1 + 12 * l + 2 * i]; };
  auto Bp = [&](int l, int i) { return (const float*)d_in[1 + 12 * l + 2 * i + 1]; };
  const float* ln_g[2] = { (const float*)d_in[37], (const float*)d_in[39] };
  const float* ln_b[2] = { (const float*)d_in[38], (const float*)d_in[40] };
  auto Es = [&](int i) { return (const int*)d_in[41 + 2 * i]; };
  auto Et = [&](int i) { return (const int*)d_in[41 + 2 * i + 1]; };
  auto En = [&](int i) { return (long long)in_sizes[41 + 2 * i]; };

  const long long HN = (long long)ROWS * FDIM;       // 4.19M floats
  float* h1 = (float*)d_ws;
  float* h2 = h1 + HN;
  float* P  = h2 + HN;                               // 48 MB total

  float* out = (float*)d_out;

  auto zgrid = [](long long n) {
    long long b = (n + 255) / 256;
    return (unsigned)(b > 65536 ? 65536 : b);
  };
  auto sgrid = [](long long n) {
    long long b = (n + 255) / 256;
    return (unsigned)(b > 262144 ? 262144 : b);
  };

  dim3 wave_rows(32, 8);                 // 8 rows per block
  const unsigned row_blocks = ROWS / 8;  // 4096
  dim3 gemm_block(32, 4);                // 4 waves (32x32 tiles) per block
  const unsigned gemm_blocks = (ROWS / 32) * (FDIM / 32) / 4;  // 1024

  // ---------------- layer 1: (2->128), no self loop ----------------
  zero_f32<<<zgrid(HN), 256, 0, stream>>>(h1, HN);
  for (int i = 1; i < NTYPES; ++i) {     // skip 'self' (i==0)
    proj_fin2<<<(unsigned)((HN + 255) / 256), 256, 0, stream>>>(x, Wp(0, i), Bp(0, i), P);
    long long tot = En(i) * BATCH * FDIM;
    scatter_add<<<sgrid(tot), 256, 0, stream>>>(P, Es(i), Et(i), h1, En(i), FDIM);
  }
  ln_tanh<<<row_blocks, wave_rows, 0, stream>>>(h1, ln_g[0], ln_b[0]);

  // ---------------- layer 2: (128->128), WMMA ----------------
  zero_f32<<<zgrid(HN), 256, 0, stream>>>(h2, HN);
  for (int i = 0; i < NTYPES; ++i) {
    gemm128_wmma_f32<<<gemm_blocks, gemm_block, 0, stream>>>(h1, Wp(1, i), Bp(1, i), P);
    long long tot = En(i) * BATCH * FDIM;
    scatter_add<<<sgrid(tot), 256, 0, stream>>>(P, Es(i), Et(i), h2, En(i), FDIM);
  }
  ln_tanh<<<row_blocks, wave_rows, 0, stream>>>(h2, ln_g[1], ln_b[1]);

  // ---------------- layer 3: (128->2) ----------------
  const long long outN = (long long)ROWS * 2;
  zero_f32<<<zgrid(outN), 256, 0, stream>>>(out, outN);
  for (int i = 0; i < NTYPES; ++i) {
    proj_fout2<<<row_blocks, wave_rows, 0, stream>>>(h2, Wp(2, i), Bp(2, i), P);
    long long tot = En(i) * BATCH * 2;
    scatter_add<<<sgrid(tot), 256, 0, stream>>>(P, Es(i), Et(i), out, En(i), 2);
  }
}